// CTCLoss_38697655337358
// MI455X (gfx1250) — compile-verified
//
#include <hip/hip_runtime.h>
#include <math.h>
#include <stdint.h>

// Problem constants (match reference setup_inputs)
#define BB 64
#define TT 160
#define CC 6625
#define LL 25
#define SS 51   // 2*LL + 1

typedef float v2f __attribute__((ext_vector_type(2)));
typedef float v8f __attribute__((ext_vector_type(8)));

#define NEG_INF (-INFINITY)

// ---------------------------------------------------------------------------
// Kernel 1: per-(b,t) row logsumexp over C.
// One block = one row of 6625 floats (26.5 KB). Stage the row into LDS via
// CDNA5 async global->LDS DMA (ASYNCcnt), then max-pass + sum-exp pass from
// LDS. Final 256-partial sum reduced with v_wmma_f32_16x16x4_f32 (ones B).
// ---------------------------------------------------------------------------
__global__ __launch_bounds__(256) void ctc_lse_kernel(const float* __restrict__ pred,
                                                      float* __restrict__ lse) {
    __shared__ float row[CC];
    __shared__ float red[256];

    const int t = threadIdx.x;
    const int r = blockIdx.x;                       // r = b*TT + time
    const float* g = pred + (size_t)r * CC;         // block-uniform -> SGPR pair

    // ---- async copy global -> LDS (coalesced b32 per lane, ASYNCcnt-tracked)
    for (int i = t; i < CC; i += 256) {
        unsigned lds_off = (unsigned)(uintptr_t)(&row[i]);   // low 32 bits = LDS byte offset
        unsigned goff    = (unsigned)(i * 4);
        asm volatile("global_load_async_to_lds_b32 %0, %1, %2"
                     :: "v"(lds_off), "v"(goff), "s"(g) : "memory");
    }
    asm volatile("s_wait_asynccnt 0" ::: "memory");
    __syncthreads();

    // ---- pass 1: row max
    float m = NEG_INF;
    for (int i = t; i < CC; i += 256) m = fmaxf(m, row[i]);
    red[t] = m;
    __syncthreads();
    for (int sft = 128; sft >= 1; sft >>= 1) {
        if (t < sft) red[t] = fmaxf(red[t], red[t + sft]);
        __syncthreads();
    }
    m = red[0];
    __syncthreads();

    // ---- pass 2: sum of exp(x - m)
    float ssum = 0.0f;
    for (int i = t; i < CC; i += 256) ssum += __expf(row[i] - m);
    red[t] = ssum;
    __syncthreads();

#if __has_builtin(__builtin_amdgcn_wmma_f32_16x16x4_f32)
    // WMMA reduction of 256 partials by wave 0 (EXEC all-ones inside the wave).
    // D = A(16x4 data) x B(4x16 ones) + C, chained over 4 chunks of 64 values.
    // A layout (ISA 7.12.2, 32-bit A 16x4): lanes 0-15 hold K=0,1; lanes 16-31 K=2,3.
    if (t < 32) {
        const int lane  = t;
        const int mrow  = lane & 15;
        const int kbase = (lane < 16) ? 0 : 2;
        v8f acc = {};
        v2f bmat; bmat[0] = 1.0f; bmat[1] = 1.0f;   // ones: layout-independent
        #pragma unroll
        for (int c = 0; c < 4; ++c) {
            v2f amat;
            amat[0] = red[c * 64 + mrow * 4 + kbase];
            amat[1] = red[c * 64 + mrow * 4 + kbase + 1];
            acc = __builtin_amdgcn_wmma_f32_16x16x4_f32(
                false, amat, false, bmat, (short)0, acc, false, false);
        }
        // every column of D identical; lanes<16 hold M=0..7, lanes>=16 hold M=8..15
        float s8 = acc[0] + acc[1] + acc[2] + acc[3] + acc[4] + acc[5] + acc[6] + acc[7];
        s8 += __shfl_xor(s8, 16, 32);
        if (lane == 0) lse[r] = m + __logf(s8);
    }
#else
    for (int sft = 128; sft >= 1; sft >>= 1) {
        if (t < sft) red[t] += red[t + sft];
        __syncthreads();
    }
    if (t == 0) lse[r] = m + __logf(red[0]);
#endif
}

// ---------------------------------------------------------------------------
// Kernel 2: gather lp[b][t][s] = pred[b,t,ext[b,s]] - lse[b,t]
// ext[b,s] = (s odd) ? targets[b][(s-1)/2] : blank(0)
// ---------------------------------------------------------------------------
__global__ __launch_bounds__(256) void ctc_gather_kernel(const float* __restrict__ pred,
                                                         const int* __restrict__ targets,
                                                         const float* __restrict__ lse,
                                                         float* __restrict__ lp) {
    const int idx = blockIdx.x * 256 + threadIdx.x;
    const int total = BB * TT * SS;
    if (idx >= total) return;
    const int s  = idx % SS;
    const int bt = idx / SS;          // b*TT + t
    const int b  = bt / TT;
    const int e  = (s & 1) ? targets[b * LL + (s >> 1)] : 0;
    lp[idx] = pred[(size_t)bt * CC + e] - lse[bt];
}

// ---------------------------------------------------------------------------
// Kernel 3: CTC forward DP. One block per batch sample, thread s owns state s.
// Serial over T with the lp load for t+1 software-pipelined across barriers.
// ---------------------------------------------------------------------------
__global__ __launch_bounds__(64) void ctc_dp_kernel(const float* __restrict__ lp,
                                                    const int* __restrict__ targets,
                                                    const int* __restrict__ tlen,
                                                    float* __restrict__ per_b) {
    __shared__ float alpha[64];
    const int b = blockIdx.x;
    const int s = threadIdx.x;
    const bool active = (s < SS);

    // skip transition allowed iff s odd, s>=3, label differs from label two back
    bool skip = false;
    if (active && (s & 1) && s >= 3) {
        const int cur = targets[b * LL + (s >> 1)];
        const int prv = targets[b * LL + (s >> 1) - 1];
        skip = (cur != 0) && (cur != prv);
    }

    const float* lpb = lp + (size_t)b * TT * SS;

    // t = 0 init: only states 0 and 1 reachable
    float a = NEG_INF;
    if (active && s <= 1) a = lpb[s];
    alpha[s] = a;
    __syncthreads();

    float nextlp = active ? lpb[SS + s] : 0.0f;   // lp for t=1, prefetched

    for (int step = 1; step < TT; ++step) {
        const float curlp = nextlp;
        if (active && (step + 1) < TT)
            nextlp = lpb[(size_t)(step + 1) * SS + s];   // hide latency over barrier

        const float a0 = alpha[s];
        const float a2 = (s >= 1) ? alpha[s - 1] : NEG_INF;
        const float a3 = skip ? alpha[s - 2] : NEG_INF;
        __syncthreads();

        const float mx = fmaxf(fmaxf(a0, a2), a3);
        float v;
        if (mx == NEG_INF) {
            v = NEG_INF;
        } else {
            v = mx + __logf(__expf(a0 - mx) + __expf(a2 - mx) + __expf(a3 - mx));
        }
        alpha[s] = active ? (v + curlp) : NEG_INF;
        __syncthreads();
    }

    if (s == 0) {
        int tl = tlen[b];
        const float l1 = alpha[2 * tl];
        const float l2 = alpha[2 * tl - 1];
        const float mx = fmaxf(l1, l2);
        float raw;
        if (mx == NEG_INF) raw = INFINITY;
        else raw = -(mx + __logf(__expf(l1 - mx) + __expf(l2 - mx)));
        const float safe = isinf(raw) ? 0.0f : raw;
        const int dv = (tl > 1) ? tl : 1;
        per_b[b] = safe / (float)dv;
    }
}

// ---------------------------------------------------------------------------
// Kernel 4: mean over batch
// ---------------------------------------------------------------------------
__global__ __launch_bounds__(64) void ctc_mean_kernel(const float* __restrict__ per_b,
                                                      float* __restrict__ out) {
    __shared__ float red[64];
    const int t = threadIdx.x;
    red[t] = per_b[t];
    __syncthreads();
    for (int sft = 32; sft >= 1; sft >>= 1) {
        if (t < sft) red[t] += red[t + sft];
        __syncthreads();
    }
    if (t == 0) out[0] = red[0] / (float)BB;
}

// ---------------------------------------------------------------------------
extern "C" void kernel_launch(void* const* d_in, const int* in_sizes, int n_in,
                              void* d_out, int out_size, void* d_ws, size_t ws_size,
                              hipStream_t stream) {
    const float* pred    = (const float*)d_in[0];   // [B,T,C] f32
    const int*   targets = (const int*)d_in[1];     // [B,L] i32
    const int*   tlen    = (const int*)d_in[2];     // [B] i32
    float*       out     = (float*)d_out;           // scalar f32

    // workspace layout
    float* lse   = (float*)d_ws;                    // B*T      = 10240
    float* lp    = lse + (BB * TT);                 // B*T*S    = 522240
    float* per_b = lp + (BB * TT * SS);             // B        = 64

    // 1) logsumexp per row (HBM-bound: 271 MB streamed once)
    ctc_lse_kernel<<<BB * TT, 256, 0, stream>>>(pred, lse);

    // 2) gather extended-label log-probs
    const int total = BB * TT * SS;
    ctc_gather_kernel<<<(total + 255) / 256, 256, 0, stream>>>(pred, targets, lse, lp);

    // 3) forward DP (serial in T, one block per sample)
    ctc_dp_kernel<<<BB, 64, 0, stream>>>(lp, targets, tlen, per_b);

    // 4) mean reduction
    ctc_mean_kernel<<<1, 64, 0, stream>>>(per_b, out);
}